// WHTConv2D_1228360646837
// MI455X (gfx1250) — compile-verified
//
#include <hip/hip_runtime.h>
#include <math.h>
#include <stdint.h>

typedef __attribute__((ext_vector_type(2))) float v2f;
typedef __attribute__((ext_vector_type(8))) float v8f;

#define HW   16384   // 128*128 pixels per (b,c) slice
#define LDW  129     // padded LDS row stride: bank = (row+col) mod 64, conflict-free

// ---------------------------------------------------------------------------
// CDNA5 async global->LDS copy (ASYNCcnt-tracked), lowers to
// global_load_async_to_lds_b32. Builtin confirmed present; param 0 is a
// non-const addrspace(1) int* per the round-2 diagnostic.
// ---------------------------------------------------------------------------
typedef __attribute__((address_space(1))) int g_i32;
typedef __attribute__((address_space(3))) int l_i32;

__device__ __forceinline__ void async_ld32(const float* g, float* l) {
#if defined(__has_builtin) && __has_builtin(__builtin_amdgcn_global_load_async_to_lds_b32)
    __builtin_amdgcn_global_load_async_to_lds_b32(
        (g_i32*)g, (l_i32*)l, 0, 0);
#else
    uint32_t loff = (uint32_t)(uintptr_t)l;
    asm volatile("global_load_async_to_lds_b32 %0, %1, off"
                 :: "v"(loff), "v"(g) : "memory");
#endif
}

__device__ __forceinline__ void wait_async0() {
#if defined(__has_builtin) && __has_builtin(__builtin_amdgcn_s_wait_asynccnt)
    __builtin_amdgcn_s_wait_asynccnt(0);
#else
    asm volatile("s_wait_asynccnt 0x0" ::: "memory");
#endif
}

// ---------------------------------------------------------------------------
// Fully register-resident 128-point FWHT (7 stages, all indices constant
// after unroll -> stays in VGPRs, 64 independent butterflies per stage).
// ---------------------------------------------------------------------------
__device__ __forceinline__ void fwht128_reg(float* r) {
    #pragma unroll
    for (int sh = 0; sh < 7; ++sh) {
        const int h = 1 << sh;
        #pragma unroll
        for (int g = 0; g < 128; g += 2 * h) {
            #pragma unroll
            for (int k = 0; k < h; ++k) {
                float a = r[g + k], b = r[g + k + h];
                r[g + k]     = a + b;
                r[g + k + h] = a - b;
            }
        }
    }
}

// ---------------------------------------------------------------------------
// FWHT2D kernels: 128 threads/block, one block per (b,c) slice.
// Thread t owns row t in registers (W pass), then column t (H pass) after a
// single conflict-free LDS transpose. PRE: *v[h,w]; POST: /16384 + x (in-place).
// ---------------------------------------------------------------------------
template <bool PRE>
__global__ __launch_bounds__(128) void k_fwht(const float* in,
                                              const float* __restrict__ aux,
                                              float* outp) {
    extern __shared__ float s[];
    const int t = threadIdx.x;
    const size_t base = (size_t)blockIdx.x * HW;   // blockIdx = b*64 + c

    float r[128];
    // Own row t: 32x b128, each lane streams its own contiguous 512B
    const float4* rowp = (const float4*)(in + base + (size_t)t * 128);
    #pragma unroll
    for (int j = 0; j < 32; ++j) {
        float4 q = rowp[j];
        r[4*j+0] = q.x; r[4*j+1] = q.y; r[4*j+2] = q.z; r[4*j+3] = q.w;
    }

    fwht128_reg(r);                                // W-direction, in registers

    #pragma unroll
    for (int i = 0; i < 128; ++i) s[t * LDW + i] = r[i];   // write row t
    __syncthreads();
    #pragma unroll
    for (int i = 0; i < 128; ++i) r[i] = s[i * LDW + t];   // read column t

    fwht128_reg(r);                                // H-direction, in registers

    if (PRE) {
        #pragma unroll
        for (int i = 0; i < 128; ++i)              // coalesced per row i
            outp[base + i * 128 + t] = r[i] * aux[i * 128 + t];
    } else {
        const float sc = 1.0f / 16384.0f;
        #pragma unroll
        for (int i = 0; i < 128; ++i)
            outp[base + i * 128 + t] = r[i] * sc + aux[base + i * 128 + t];
    }
}

// ---------------------------------------------------------------------------
// Channel mix + soft-threshold, in place on 32-pixel column blocks.
//   D[o,pix] = soft_threshold( sum_c Wm[o,c] * F[c,pix], T[pix] )
// 8 waves = 4 M-tiles x 2 N-tiles; K swept 4-wide via V_WMMA_F32_16X16X4_F32.
// Activation tile loaded transposed into LDS via async global->LDS gather.
// ---------------------------------------------------------------------------
__global__ __launch_bounds__(256) void k_cmix(float* __restrict__ Fall,
                                              const float* __restrict__ Wm,
                                              const float* __restrict__ T) {
    __shared__ float lds[32 * 66];                 // [pixel][channel], pad 64->66
    const int t = threadIdx.x;
    const int b       = blockIdx.x >> 9;           // 512 pixel-blocks per batch
    const int pixbase = (blockIdx.x & 511) << 5;   // 32 pixels per block
    float* F = Fall + (size_t)b * 64 * HW;

    // Async transposed gather: global (channel-major) -> LDS (pixel-major)
    #pragma unroll
    for (int j = 0; j < 8; ++j) {
        int i  = t + (j << 8);                     // 0..2047
        int c  = i >> 5;
        int px = i & 31;
        async_ld32(&F[(size_t)c * HW + pixbase + px], &lds[px * 66 + c]);
    }
    wait_async0();
    __syncthreads();

    const int lane  = t & 31;
    const int wid   = t >> 5;
    const int mtile = wid & 3;
    const int nt    = wid >> 2;
    const int m     = lane & 15;
    const int half  = lane >> 4;

    // A (16x4 f32 / step): VGPR0 K=0|K=2 across half-waves, VGPR1 K=1|K=3
    v2f A[16];
    const float* wrow = Wm + (mtile * 16 + m) * 64 + 2 * half;
    #pragma unroll
    for (int k = 0; k < 16; ++k) {
        A[k].x = wrow[4 * k + 0];
        A[k].y = wrow[4 * k + 1];
    }

    // B (4x16 f32 / step): N = lane&15; VGPR0 K=0|K=2, VGPR1 K=1|K=3
    v8f acc = {};
    const float* bcol = lds + (nt * 16 + m) * 66 + 2 * half;
    #pragma unroll
    for (int k = 0; k < 16; ++k) {
        v2f bb;
        bb.x = bcol[4 * k + 0];
        bb.y = bcol[4 * k + 1];
        acc = __builtin_amdgcn_wmma_f32_16x16x4_f32(
            false, A[k], false, bb, (short)0, acc, false, false);
    }

    // D: VGPR r holds rows M=r (lanes 0-15) and M=r+8 (lanes 16-31)
    const int   pix = pixbase + nt * 16 + m;
    const float tr  = fmaxf(T[pix], 0.0f);         // relu(T)
    #pragma unroll
    for (int r = 0; r < 8; ++r) {
        int   o   = mtile * 16 + r + 8 * half;
        float val = acc[r];
        float mag = fmaxf(fabsf(val) - tr, 0.0f);
        F[(size_t)o * HW + pix] = copysignf(mag, val);
    }
}

extern "C" void kernel_launch(void* const* d_in, const int* in_sizes, int n_in,
                              void* d_out, int out_size, void* d_ws, size_t ws_size,
                              hipStream_t stream) {
    const float* x  = (const float*)d_in[0];   // (32,64,128,128)
    const float* Wm = (const float*)d_in[1];   // (64,64)
    const float* v  = (const float*)d_in[2];   // (128,128)
    const float* T  = (const float*)d_in[3];   // (128,128)
    float* out = (float*)d_out;

    const int    BC       = 32 * 64;                    // one block per (b,c) slice
    const size_t ldsBytes = (size_t)128 * LDW * sizeof(float);

    k_fwht<true>  <<<BC,       128, ldsBytes, stream>>>(x, v, out);
    k_cmix        <<<32 * 512, 256, 0,        stream>>>(out, Wm, T);
    k_fwht<false> <<<BC,       128, ldsBytes, stream>>>(out, x, out);
}